// _Expert_78632261255734
// MI455X (gfx1250) — compile-verified
//
#include <hip/hip_runtime.h>
#include <math.h>

// ---------------------------------------------------------------------------
// MoE expert FFN for MI455X (gfx1250, wave32).
// Memory-bound (603 MB @ 23.3 TB/s ~ 26us >> 137 GFLOP of fp32 WMMA), so we
// keep native fp32 and use V_WMMA_F32_16X16X4_F32. L2 (192 MB) holds each
// expert's full working set, so direct global loads get full M x N reuse.
// ---------------------------------------------------------------------------

typedef float v2f __attribute__((ext_vector_type(2)));
typedef float v8f __attribute__((ext_vector_type(8)));

__device__ __forceinline__ float gelu_exact(float x) {
    // 0.5 * x * (1 + erf(x / sqrt(2)))  -- matches jax.nn.gelu(approximate=False)
    return 0.5f * x * (1.0f + erff(x * 0.70710678118654752440f));
}

// C[off+row, 0..N) = act( A[off+row, :K] . B_e[col, :K] + bias_e[col] )
//  A    : [T x K] row-major, rows addressed by global token index
//  B    : [E x N x K] row-major (so the GEMM is A . B^T, matching X @ W.T)
//  bias : [E x N]
//  C    : [T x N] row-major
// One wave computes a 64x64 tile as 4x4 WMMA 16x16 tiles; 4 waves -> 128x128
// per block. grid = (N/128, ceil(T/128), E); blocks past the expert's segment
// exit immediately.
template <bool GELU>
__global__ __launch_bounds__(128) void moe_gemm_wmma_f32(
    const float* __restrict__ A,
    const float* __restrict__ B,
    const float* __restrict__ bias,
    float* __restrict__ C,
    const int* __restrict__ counts,   // fwd_expert_count, [E]
    int E, int K, int N)
{
    const int e = blockIdx.z;

    // Segment offset/count from device-side counts (uniform -> scalar loads).
    int off = 0;
    for (int i = 0; i < e; ++i) off += counts[i];
    const int cnt = counts[e];

    const int mbase = blockIdx.y * 128;
    if (mbase >= cnt) return;   // nothing for this expert in this row-tile

    const int tid  = threadIdx.x;
    const int wave = tid >> 5;
    const int lane = tid & 31;
    const int laneM    = lane & 15;          // row-in-16 for A/B fragment loads
    const int laneKoff = (lane >> 4) * 2;    // lanes 0-15 hold K=0,1; 16-31 hold K=2,3

    const int wm    = (wave >> 1) * 64;      // wave's 64x64 tile inside 128x128
    const int wn    = (wave & 1) * 64;
    const int nbase = blockIdx.x * 128 + wn;

    const float* __restrict__ Bexp = B + (size_t)e * (size_t)N * (size_t)K;

    // Fragment base pointers (A rows clamped into segment so OOB rows read
    // in-bounds garbage; their stores are guarded below).
    const float* ap[4];
    const float* bp[4];
#pragma unroll
    for (int mt = 0; mt < 4; ++mt) {
        int r = mbase + wm + mt * 16 + laneM;
        r = (r < cnt) ? r : (cnt - 1);
        ap[mt] = A + (size_t)(off + r) * (size_t)K + laneKoff;
    }
#pragma unroll
    for (int nt = 0; nt < 4; ++nt) {
        int c = nbase + nt * 16 + laneM;
        bp[nt] = Bexp + (size_t)c * (size_t)K + laneKoff;
    }

    v8f acc[4][4];
#pragma unroll
    for (int mt = 0; mt < 4; ++mt)
#pragma unroll
        for (int nt = 0; nt < 4; ++nt)
            acc[mt][nt] = v8f{};

    // K loop: 8 x b64 loads feed 16 WMMAs per step (32 KFLOP per 2 KB).
#pragma unroll 2
    for (int k = 0; k < K; k += 4) {
        v2f a[4], b[4];
#pragma unroll
        for (int mt = 0; mt < 4; ++mt)
            a[mt] = *(const v2f*)(ap[mt] + k);
#pragma unroll
        for (int nt = 0; nt < 4; ++nt)
            b[nt] = *(const v2f*)(bp[nt] + k);

#pragma unroll
        for (int mt = 0; mt < 4; ++mt)
#pragma unroll
            for (int nt = 0; nt < 4; ++nt)
                acc[mt][nt] = __builtin_amdgcn_wmma_f32_16x16x4_f32(
                    /*neg_a=*/false, a[mt],
                    /*neg_b=*/false, b[nt],
                    /*c_mod=*/(short)0, acc[mt][nt],
                    /*reuse_a=*/false, /*reuse_b=*/false);
    }

    // Epilogue. C/D layout: VGPR r -> row r + 8*(lane>=16), col = lane&15.
    const int rOff = 8 * (lane >> 4);
#pragma unroll
    for (int nt = 0; nt < 4; ++nt) {
        const int col = nbase + nt * 16 + laneM;
        const float bv = bias[(size_t)e * N + col];
#pragma unroll
        for (int mt = 0; mt < 4; ++mt) {
#pragma unroll
            for (int r = 0; r < 8; ++r) {
                const int row = mbase + wm + mt * 16 + r + rOff;
                if (row < cnt) {
                    float v = acc[mt][nt][r] + bv;
                    if (GELU) v = gelu_exact(v);
                    C[(size_t)(off + row) * (size_t)N + col] = v;
                }
            }
        }
    }
}

extern "C" void kernel_launch(void* const* d_in, const int* in_sizes, int n_in,
                              void* d_out, int out_size, void* d_ws, size_t ws_size,
                              hipStream_t stream) {
    const float* inp = (const float*)d_in[0];   // [T, D]
    const float* w1  = (const float*)d_in[1];   // [E, H, D]
    const float* b1  = (const float*)d_in[2];   // [E, H]
    const float* w2  = (const float*)d_in[3];   // [E, D, H]
    const float* b2  = (const float*)d_in[4];   // [E, D]
    const int*   cnt = (const int*)d_in[5];     // [E]
    float*       out = (float*)d_out;           // [T, D]
    float*       h   = (float*)d_ws;            // [T, H] intermediate (T*H*4 bytes)

    const int E = in_sizes[5];
    const int H = in_sizes[2] / E;
    const int D = in_sizes[4] / E;
    const int T = in_sizes[0] / D;

    dim3 blk(128, 1, 1);
    // GEMM1: h = gelu(inp . w1^T + b1)   M<=T(512/expert), N=H, K=D
    dim3 g1(H / 128, (T + 127) / 128, E);
    moe_gemm_wmma_f32<true><<<g1, blk, 0, stream>>>(inp, w1, b1, h, cnt, E, D, H);
    // GEMM2: out = h . w2^T + b2         N=D, K=H
    dim3 g2(D / 128, (T + 127) / 128, E);
    moe_gemm_wmma_f32<false><<<g2, blk, 0, stream>>>(h, w2, b2, out, cnt, E, H, D);
}